// MaskCNN1_13907104104939
// MI455X (gfx1250) — compile-verified
//
#include <hip/hip_runtime.h>
#include <hip/hip_bf16.h>

typedef float v4f __attribute__((ext_vector_type(4)));

// Grid: (C*H / 8, B). One block = 8 consecutive W-rows of one batch.
// 256 threads x 4 floats = one 1024-float row per "slice"; the 8 rows share
// both `length` (same batch) and per-lane `pos`, so ONE mask classification
// covers all 8 load/store pairs. 8 b128 NT loads issued back-to-back then
// 8 NT stores -> 128B in flight per lane, deep MLP for the 23.3 TB/s stream.
// Masked region writes zeros WITHOUT loading x. NT hints keep the one-touch
// 640MB stream out of L2.
#define ROWS_PER_BLOCK 8

__global__ __launch_bounds__(256) void maskcnn_kernel(
    const float* __restrict__ x,
    const float* __restrict__ percents,
    float* __restrict__ out,
    int rows_per_batch)   // C*H (multiple of ROWS_PER_BLOCK)
{
    constexpr int W = 1024;

    const int b       = blockIdx.y;                      // batch (scalar)
    const int rowBase = blockIdx.x * ROWS_PER_BLOCK;     // first row in batch
    const int length  = (int)((float)W * percents[b]);   // uniform: s_load path

    const int pos = (int)threadIdx.x * 4;                // 0,4,...,1020
    const size_t base0 =
        ((size_t)b * (size_t)rows_per_batch + (size_t)rowBase) * W + (size_t)pos;

    v4f v[ROWS_PER_BLOCK];

    if (pos + 4 <= length) {
        // fully valid for all 8 rows: straight streaming copies
#pragma unroll
        for (int r = 0; r < ROWS_PER_BLOCK; ++r)
            v[r] = __builtin_nontemporal_load((const v4f*)(x + base0 + (size_t)r * W));
    } else if (pos >= length) {
        // fully masked for all 8 rows: zeros, no loads at all
#pragma unroll
        for (int r = 0; r < ROWS_PER_BLOCK; ++r)
            v[r] = v4f{0.0f, 0.0f, 0.0f, 0.0f};
    } else {
        // boundary vector (at most one wave per block takes this path)
#pragma unroll
        for (int r = 0; r < ROWS_PER_BLOCK; ++r) {
            v[r] = __builtin_nontemporal_load((const v4f*)(x + base0 + (size_t)r * W));
#pragma unroll
            for (int i = 0; i < 4; ++i)
                if (pos + i >= length) v[r][i] = 0.0f;
        }
    }

#pragma unroll
    for (int r = 0; r < ROWS_PER_BLOCK; ++r)
        __builtin_nontemporal_store(v[r], (v4f*)(out + base0 + (size_t)r * W));
}

extern "C" void kernel_launch(void* const* d_in, const int* in_sizes, int n_in,
                              void* d_out, int out_size, void* d_ws, size_t ws_size,
                              hipStream_t stream) {
    const float* x        = (const float*)d_in[0];
    const float* percents = (const float*)d_in[1];
    float*       out      = (float*)d_out;

    constexpr int W = 1024;
    const int total = in_sizes[0];          // B*C*H*W = 83,886,080
    const int B     = in_sizes[1];          // 16
    const int rows  = total / W;            // B*C*H = 81,920
    const int rows_per_batch = rows / B;    // C*H   = 5,120 (divisible by 8)

    dim3 grid((unsigned)(rows_per_batch / ROWS_PER_BLOCK), (unsigned)B, 1);
    maskcnn_kernel<<<grid, 256, 0, stream>>>(x, percents, out, rows_per_batch);
}